// MetaNetLinearizedModel_68504728371609
// MI455X (gfx1250) — compile-verified
//
#include <hip/hip_runtime.h>
#include <hip/hip_bf16.h>

// ---------------------------------------------------------------------------
// MetaNet linearized model, MI455X (gfx1250, wave32, WMMA).
//
// All heavy math is skinny fp32 GEMM  C[32,N] = A[32,K] @ B[K,N], executed
// with V_WMMA_F32_16X16X4_F32. Each wave computes a 32x(16*NT) C tile
// (NT=2 normally -> 4 wmma per 4-K step, A fragments reused across N tiles).
// Split-K partials to workspace + deterministic fused reduce.
// Memory-bound problem (~210 MB streamed, 3.4 GFLOP): fp32 end-to-end.
// ---------------------------------------------------------------------------

typedef __attribute__((ext_vector_type(2))) float v2f;
typedef __attribute__((ext_vector_type(8))) float v8f;

#define BATCH 32
#define DD 768
#define HH 3072
#define TT 8
#define METAH 192
#define NCOEF 48   // T * P_BLOCKS

// ---------------- patch mean:  xbar[b, c*256+i*16+j] = mean over 14x14 patches
__global__ void patch_mean_kernel(const float* __restrict__ x, float* __restrict__ xbar) {
    int idx = blockIdx.x * blockDim.x + threadIdx.x;   // 32*768
    if (idx >= BATCH * DD) return;
    int b = idx / DD, d = idx % DD;
    int ch = d >> 8, rem = d & 255, i = rem >> 4, j = rem & 15;
    const float* base = x + (((size_t)b * 3 + ch) * 224) * 224;
    float s = 0.f;
    #pragma unroll 2
    for (int ph = 0; ph < 14; ++ph) {
        const float* row = base + (size_t)(ph * 16 + i) * 224 + j;
        #pragma unroll
        for (int pw = 0; pw < 14; ++pw) s += row[pw * 16];
    }
    xbar[idx] = s * (1.0f / 196.0f);
}

// ---------------- split-K WMMA GEMM partial:
// grid.x = N/(16*NT) tiles, grid.y = k-split index. One wave per block.
// Writes a 32 x (16*NT) partial tile into P[ks][32][N].
template <int NT>
__global__ __launch_bounds__(32)
void gemm32_part_kernel(const float* __restrict__ A,   // [32,K] row-major
                        const float* __restrict__ B,   // [K,N] row-major
                        float* __restrict__ P,         // partial base [KS][32][N]
                        int K, int N, int kChunk) {
    const int lane  = threadIdx.x;        // 0..31
    const int ntile = blockIdx.x;
    const int ks    = blockIdx.y;
    const int k0 = ks * kChunk;
    int k1 = k0 + kChunk; if (k1 > K) k1 = K;

    const int half = lane >> 4;           // 0: K+0,1   1: K+2,3
    const int l    = lane & 15;
    const int kOff = half * 2;
    const int colBase = ntile * (16 * NT);

    // strength-reduced streaming pointers (no per-iter 64-bit mul)
    const float* pA0 = A + (size_t)l * K        + k0 + kOff;   // M rows 0..15
    const float* pA1 = A + (size_t)(l + 16) * K + k0 + kOff;   // M rows 16..31
    const float* pB  = B + (size_t)(k0 + kOff) * N + colBase + l;
    const size_t bStep = (size_t)4 * N;

    v8f acc[NT][2];
    #pragma unroll
    for (int j = 0; j < NT; ++j) { acc[j][0] = (v8f){}; acc[j][1] = (v8f){}; }

    #pragma unroll 2
    for (int k = k0; k < k1; k += 4) {
        v2f a0 = *(const v2f*)pA0;  pA0 += 4;
        v2f a1 = *(const v2f*)pA1;  pA1 += 4;
        v2f bf[NT];
        #pragma unroll
        for (int j = 0; j < NT; ++j) {
            bf[j].x = pB[j * 16];
            bf[j].y = pB[j * 16 + N];
        }
        pB += bStep;
        #pragma unroll
        for (int j = 0; j < NT; ++j) {
            acc[j][0] = __builtin_amdgcn_wmma_f32_16x16x4_f32(false, a0, false, bf[j], (short)0, acc[j][0], false, false);
            acc[j][1] = __builtin_amdgcn_wmma_f32_16x16x4_f32(false, a1, false, bf[j], (short)0, acc[j][1], false, false);
        }
    }

    // C layout: lanes 0-15 hold cols, VGPR r -> row r; lanes 16-31 -> row 8+r.
    float* Pt = P + ((size_t)ks * BATCH) * N + colBase + l;
    const int rbase = half * 8;
    #pragma unroll
    for (int j = 0; j < NT; ++j) {
        #pragma unroll
        for (int r = 0; r < 8; ++r) {
            Pt[(size_t)(rbase + r) * N      + j * 16] = acc[j][0][r];
            Pt[(size_t)(rbase + r + 16) * N + j * 16] = acc[j][1][r];
        }
    }
}

// ---------------- deterministic split-K reduce with fused epilogue
__global__ void reduce_combine_kernel(float* __restrict__ Cout,
                                      const float* __restrict__ P,
                                      int KS, int N,
                                      const float* __restrict__ bias,     // [N] or null
                                      const float* __restrict__ add,      // [32,N] residual or null
                                      const float* __restrict__ maskSrc,  // zero where <=0 (relu') or null
                                      float* __restrict__ Crelu,          // relu copy out or null
                                      int reluMain) {
    int idx = blockIdx.x * blockDim.x + threadIdx.x;
    int total = BATCH * N;
    if (idx >= total) return;
    int n = idx % N;
    float v = 0.f;
    for (int s = 0; s < KS; ++s) v += P[(size_t)s * total + idx];
    if (bias)    v += bias[n];
    if (add)     v += add[idx];
    if (maskSrc) v = (maskSrc[idx] > 0.f) ? v : 0.f;
    if (reluMain) v = fmaxf(v, 0.f);
    Cout[idx] = v;
    if (Crelu) Crelu[idx] = fmaxf(v, 0.f);
}

// ---------------- extract coef slices: cs[s][b][t] = coefs[b, t*6+s]
__global__ void coef_slices_kernel(const float* __restrict__ coefs, float* __restrict__ cs) {
    int idx = blockIdx.x * blockDim.x + threadIdx.x;   // 6*32*8
    if (idx >= 6 * BATCH * TT) return;
    int s = idx / (BATCH * TT), r = idx % (BATCH * TT), b = r / TT, t = r % TT;
    cs[(size_t)s * BATCH * TT + b * TT + t] = coefs[b * NCOEF + t * 6 + s];
}

// ---------------- build scaled-A: dst[b, t*K0+k] = coefs[b,t,slice] * src[b,k]
__global__ void scale_expand_kernel(const float* __restrict__ src, const float* __restrict__ coefs,
                                    float* __restrict__ dst, int K0, int sliceIdx) {
    int idx = blockIdx.x * blockDim.x + threadIdx.x;   // 32*8*K0
    int total = BATCH * TT * K0;
    if (idx >= total) return;
    int b = idx / (TT * K0);
    int r = idx % (TT * K0);
    int t = r / K0;
    int k = r % K0;
    dst[(size_t)b * (TT * K0) + r] = coefs[b * NCOEF + t * 6 + sliceIdx] * src[(size_t)b * K0 + k];
}

// ---------------------------------------------------------------------------

static inline int ceil_div(int a, int b) { return (a + b - 1) / b; }

extern "C" void kernel_launch(void* const* d_in, const int* in_sizes, int n_in,
                              void* d_out, int out_size, void* d_ws, size_t ws_size,
                              hipStream_t stream) {
    const float* x   = (const float*)d_in[0];
    const float* Wp  = (const float*)d_in[1];
    const float* bp  = (const float*)d_in[2];
    const float* W1  = (const float*)d_in[3];
    const float* b1  = (const float*)d_in[4];
    const float* W2  = (const float*)d_in[5];
    const float* b2  = (const float*)d_in[6];
    const float* dWp = (const float*)d_in[7];   // [8,768,768]  -> view [6144,768]
    const float* dbp = (const float*)d_in[8];   // [8,768]
    const float* dW1 = (const float*)d_in[9];   // [8,768,3072] -> view [6144,3072]
    const float* db1 = (const float*)d_in[10];  // [8,3072]
    const float* dW2 = (const float*)d_in[11];  // [8,3072,768] -> view [24576,768]
    const float* db2 = (const float*)d_in[12];  // [8,768]
    const float* mW1 = (const float*)d_in[13];
    const float* mb1 = (const float*)d_in[14];
    const float* mW2 = (const float*)d_in[15];
    const float* mb2 = (const float*)d_in[16];
    float* out = (float*)d_out;

    // workspace layout (floats)
    float* ws = (float*)d_ws;
    size_t off = 0;
    float* xbar    = ws + off; off += BATCH * DD;        // 24576
    float* fbuf    = ws + off; off += BATCH * DD;
    float* ubuf    = ws + off; off += BATCH * HH;        // 98304
    float* gbuf    = ws + off; off += BATCH * HH;
    float* outb    = ws + off; off += BATCH * DD;
    float* hmeta   = ws + off; off += BATCH * METAH;
    float* coefs   = ws + off; off += BATCH * NCOEF;
    float* cslice  = ws + off; off += 6 * BATCH * TT;
    float* dgbuf   = ws + off; off += BATCH * HH;
    float* dfbuf   = ws + off; off += BATCH * DD;
    float* Ascaled = ws + off; off += (size_t)BATCH * TT * HH;   // 786432 (max)
    float* P       = ws + off;                                   // partials, ~3.25M floats max

    dim3 blk32(32);

    // 1) patch means
    patch_mean_kernel<<<ceil_div(BATCH * DD, 256), 256, 0, stream>>>(x, xbar);

    // 2) f = xbar @ Wp + bp      K=768, N=768, KS=16 (chunk 48)
    gemm32_part_kernel<2><<<dim3(DD / 32, 16), blk32, 0, stream>>>(xbar, Wp, P, DD, DD, 48);
    reduce_combine_kernel<<<ceil_div(BATCH * DD, 256), 256, 0, stream>>>(
        fbuf, P, 16, DD, bp, nullptr, nullptr, nullptr, 0);

    // 3) u = f @ W1 + b1, g = relu(u)   K=768, N=3072, KS=8 (chunk 96)
    gemm32_part_kernel<2><<<dim3(HH / 32, 8), blk32, 0, stream>>>(fbuf, W1, P, DD, HH, 96);
    reduce_combine_kernel<<<ceil_div(BATCH * HH, 256), 256, 0, stream>>>(
        ubuf, P, 8, HH, b1, nullptr, nullptr, gbuf, 0);

    // 4) out_base = g @ W2 + b2   K=3072, N=768, KS=32 (chunk 96)
    gemm32_part_kernel<2><<<dim3(DD / 32, 32), blk32, 0, stream>>>(gbuf, W2, P, HH, DD, 96);
    reduce_combine_kernel<<<ceil_div(BATCH * DD, 256), 256, 0, stream>>>(
        outb, P, 32, DD, b2, nullptr, nullptr, nullptr, 0);

    // 5) hmeta = relu(out_base @ mW1 + mb1)   K=768, N=192, KS=16 (chunk 48)
    gemm32_part_kernel<2><<<dim3(METAH / 32, 16), blk32, 0, stream>>>(outb, mW1, P, DD, METAH, 48);
    reduce_combine_kernel<<<ceil_div(BATCH * METAH, 256), 256, 0, stream>>>(
        hmeta, P, 16, METAH, mb1, nullptr, nullptr, nullptr, 1);

    // 6) coefs = hmeta @ mW2 + mb2   K=192, N=48 (not 32-divisible -> NT=1), KS=4
    gemm32_part_kernel<1><<<dim3(NCOEF / 16, 4), blk32, 0, stream>>>(hmeta, mW2, P, METAH, NCOEF, 48);
    reduce_combine_kernel<<<ceil_div(BATCH * NCOEF, 256), 256, 0, stream>>>(
        coefs, P, 4, NCOEF, mb2, nullptr, nullptr, nullptr, 0);

    // 7) per-block coef slices [6][32][8]
    coef_slices_kernel<<<ceil_div(6 * BATCH * TT, 256), 256, 0, stream>>>(coefs, cslice);

    // 8) df = A0s @ dWp_view + cslice1 @ dbp
    //    A0s[b,t*768+k] = c[b,t,0]*xbar[b,k];  K=6144 (KS=32, chunk 192) + K=8 (1 slot)
    scale_expand_kernel<<<ceil_div(BATCH * TT * DD, 256), 256, 0, stream>>>(xbar, coefs, Ascaled, DD, 0);
    gemm32_part_kernel<2><<<dim3(DD / 32, 32), blk32, 0, stream>>>(Ascaled, dWp, P, TT * DD, DD, 192);
    gemm32_part_kernel<2><<<dim3(DD / 32, 1), blk32, 0, stream>>>(
        cslice + 1 * BATCH * TT, dbp, P + (size_t)32 * BATCH * DD, TT, DD, 8);
    reduce_combine_kernel<<<ceil_div(BATCH * DD, 256), 256, 0, stream>>>(
        dfbuf, P, 33, DD, nullptr, nullptr, nullptr, nullptr, 0);

    // 9) dg = (u>0) * (df @ W1 + A1s @ dW1_view + cslice3 @ db1)
    //    slots: df@W1 KS=8 (chunk 96) | A1s@dW1 K=6144 KS=24 (chunk 256) | K=8 1 slot
    scale_expand_kernel<<<ceil_div(BATCH * TT * DD, 256), 256, 0, stream>>>(fbuf, coefs, Ascaled, DD, 2);
    gemm32_part_kernel<2><<<dim3(HH / 32, 8), blk32, 0, stream>>>(dfbuf, W1, P, DD, HH, 96);
    gemm32_part_kernel<2><<<dim3(HH / 32, 24), blk32, 0, stream>>>(
        Ascaled, dW1, P + (size_t)8 * BATCH * HH, TT * DD, HH, 256);
    gemm32_part_kernel<2><<<dim3(HH / 32, 1), blk32, 0, stream>>>(
        cslice + 3 * BATCH * TT, db1, P + (size_t)32 * BATCH * HH, TT, HH, 8);
    reduce_combine_kernel<<<ceil_div(BATCH * HH, 256), 256, 0, stream>>>(
        dgbuf, P, 33, HH, nullptr, nullptr, ubuf, nullptr, 0);

    // 10) result = out_base + dg @ W2 + A2s @ dW2_view + cslice5 @ db2
    //     slots: dg@W2 K=3072 KS=16 (chunk 192) | A2s@dW2 K=24576 KS=48 (chunk 512) | K=8 1 slot
    scale_expand_kernel<<<ceil_div(BATCH * TT * HH, 256), 256, 0, stream>>>(gbuf, coefs, Ascaled, HH, 4);
    gemm32_part_kernel<2><<<dim3(DD / 32, 16), blk32, 0, stream>>>(dgbuf, W2, P, HH, DD, 192);
    gemm32_part_kernel<2><<<dim3(DD / 32, 48), blk32, 0, stream>>>(
        Ascaled, dW2, P + (size_t)16 * BATCH * DD, TT * HH, DD, 512);
    gemm32_part_kernel<2><<<dim3(DD / 32, 1), blk32, 0, stream>>>(
        cslice + 5 * BATCH * TT, db2, P + (size_t)64 * BATCH * DD, TT, DD, 8);
    reduce_combine_kernel<<<ceil_div(BATCH * DD, 256), 256, 0, stream>>>(
        out, P, 65, DD, nullptr, outb, nullptr, nullptr, 0);
}